// GedLayer_89438398972060
// MI455X (gfx1250) — compile-verified
//
#include <hip/hip_runtime.h>

#define NN 95
#define NP 96
#define NLAB 8
#define ITERS 10
#define BLOCK 256

typedef __attribute__((ext_vector_type(2))) float v2f;
typedef __attribute__((ext_vector_type(8))) float v8f;

__device__ __forceinline__ float relu_f(float x) { return fmaxf(x, 0.0f); }

__device__ __forceinline__ float block_sum(float v, float* red, int tid) {
    red[tid] = v;
    __syncthreads();
    #pragma unroll
    for (int s = BLOCK / 2; s > 0; s >>= 1) {
        if (tid < s) red[tid] += red[tid + s];
        __syncthreads();
    }
    float r = red[0];
    __syncthreads();
    return r;
}

__global__ __launch_bounds__(BLOCK, 1)
void ged_kernel(const int* __restrict__ A1, const int* __restrict__ A2,
                const int* __restrict__ l1, const int* __restrict__ l2,
                const float* __restrict__ nw, const float* __restrict__ ew,
                float* __restrict__ out)
{
    __shared__ float S[NP * NP];          // Sinkhorn matrix (stays resident)
    __shared__ float T1[NP * NP];         // intermediate B_q @ S^T
    __shared__ unsigned char a1s[NP * NP]; // padded A1 labels (byte-packed)
    __shared__ unsigned char a2s[NP * NP]; // padded A2 labels
    __shared__ float NCx[9 * 9];          // node cost table (incl. ins/del row/col)
    __shared__ float EC[4 * 4];           // edge cost table (incl. ins/del row/col)
    __shared__ int l1e[NP], l2e[NP];
    __shared__ float rsum[NP];
    __shared__ float red[BLOCK];

    const int tid  = threadIdx.x;
    const int lane = tid & 31;
    const int wave = tid >> 5;

    // ---- Build cost tables from weighs (relu + symmetric triu packing) ----
    if (tid < 81) {
        int a = tid / 9, b = tid % 9;
        float v;
        if (a == b)                      v = 0.0f;
        else if (a == NLAB || b == NLAB) v = relu_f(nw[28]);          // nodeInsDel
        else {
            int r = min(a, b), c = max(a, b);
            int idx = r * 7 - (r * (r - 1)) / 2 + (c - r - 1);        // triu(8,1) index
            v = relu_f(nw[idx]);
        }
        NCx[a * 9 + b] = v;
    } else if (tid < 97) {
        int e = tid - 81;
        int r = e / 4, c = e % 4;
        float v;
        if (r == c)              v = 0.0f;
        else if (r == 0 || c == 0) v = relu_f(ew[3]);                 // edgeInsDel
        else {
            int rr = min(r, c) - 1, cc = max(r, c) - 1;
            int idx = rr * 2 - (rr * (rr - 1)) / 2 + (cc - rr - 1);   // triu(3,1) index
            v = relu_f(ew[idx]);
        }
        EC[r * 4 + c] = v;
    }
    if (tid < NP) {
        l1e[tid] = (tid < NN) ? l1[tid] : NLAB;
        l2e[tid] = (tid < NN) ? l2[tid] : NLAB;
    }
    for (int idx = tid; idx < NP * NP; idx += BLOCK) {
        int i = idx / NP, j = idx % NP;
        a1s[idx] = (i < NN && j < NN) ? (unsigned char)A1[i * NN + j] : 0;
        a2s[idx] = (i < NN && j < NN) ? (unsigned char)A2[i * NN + j] : 0;
    }
    __syncthreads();

    // ---- S = exp(-0.5 * cvec) ----
    for (int idx = tid; idx < NP * NP; idx += BLOCK) {
        int i = idx / NP, k = idx % NP;
        S[idx] = __expf(-0.5f * NCx[l1e[i] * 9 + l2e[k]]);
    }
    __syncthreads();

    // ---- Sinkhorn: 10 x (row-normalize, col-normalize) ----
    for (int it = 0; it < ITERS; ++it) {
        if (tid < NP) {
            float s = 0.0f;
            for (int k = 0; k < NP; ++k) s += S[tid * NP + k];
            rsum[tid] = s;
        }
        __syncthreads();
        for (int idx = tid; idx < NP * NP; idx += BLOCK) S[idx] /= rsum[idx / NP];
        __syncthreads();
        if (tid < NP) {
            float s = 0.0f;
            for (int r = 0; r < NP; ++r) s += S[r * NP + tid];
            rsum[tid] = s;
        }
        __syncthreads();
        for (int idx = tid; idx < NP * NP; idx += BLOCK) S[idx] /= rsum[idx % NP];
        __syncthreads();
    }

    // ---- Quadratic term via fp32 WMMA:
    //   vCv = sum_q sum_{i,j} EC[a1[i,j], q] * (S @ B_q @ S^T)[j, i]
    float vcv_local = 0.0f;
    const int ln = lane & 15;   // row/col within fragment
    const int hi = lane >> 4;   // K-half selector (K=0,1 vs K=2,3)

    for (int q = 0; q < 4; ++q) {
        // Stage 1: T1[k,i] = sum_l (a2[k,l]==q) * S[i,l]   (A built on the fly, B = S^T)
        for (int tile = wave; tile < 36; tile += 8) {
            int tm = (tile / 6) * 16, tn = (tile % 6) * 16;
            v8f acc = {};
            for (int k0 = 0; k0 < NP; k0 += 4) {
                int kk = k0 + (hi << 1);
                v2f a, b;
                a.x = (a2s[(tm + ln) * NP + kk    ] == q) ? 1.0f : 0.0f;
                a.y = (a2s[(tm + ln) * NP + kk + 1] == q) ? 1.0f : 0.0f;
                b.x = S[(tn + ln) * NP + kk];          // transposed read of S
                b.y = S[(tn + ln) * NP + kk + 1];
                acc = __builtin_amdgcn_wmma_f32_16x16x4_f32(
                          false, a, false, b, (short)0, acc, false, false);
            }
            #pragma unroll
            for (int g = 0; g < 8; ++g)
                T1[(tm + g + 8 * hi) * NP + tn + ln] = acc[g];
        }
        __syncthreads();

        // Stage 2: G[j,i] = sum_k S[j,k] * T1[k,i]; reduce against EC[a1[i,j],q] in-register
        for (int tile = wave; tile < 36; tile += 8) {
            int tm = (tile / 6) * 16, tn = (tile % 6) * 16;
            v8f acc = {};
            for (int k0 = 0; k0 < NP; k0 += 4) {
                int kk = k0 + (hi << 1);
                v2f a, b;
                a.x = S[(tm + ln) * NP + kk];
                a.y = S[(tm + ln) * NP + kk + 1];
                b.x = T1[kk * NP + tn + ln];
                b.y = T1[(kk + 1) * NP + tn + ln];
                acc = __builtin_amdgcn_wmma_f32_16x16x4_f32(
                          false, a, false, b, (short)0, acc, false, false);
            }
            #pragma unroll
            for (int g = 0; g < 8; ++g) {
                int j = tm + g + 8 * hi;
                int i = tn + ln;
                int p = a1s[i * NP + j];
                vcv_local += EC[p * 4 + q] * acc[g];
            }
        }
        __syncthreads();
    }
    float vCv = block_sum(vcv_local, red, tid);

    // ---- Linear term: cvec . v ----
    float cv_local = 0.0f;
    for (int idx = tid; idx < NP * NP; idx += BLOCK) {
        int i = idx / NP, k = idx % NP;
        cv_local += S[idx] * NCx[l1e[i] * 9 + l2e[k]];
    }
    float cv = block_sum(cv_local, red, tid);

    if (tid == 0) out[0] = 0.5f * vCv + cv;
}

extern "C" void kernel_launch(void* const* d_in, const int* in_sizes, int n_in,
                              void* d_out, int out_size, void* d_ws, size_t ws_size,
                              hipStream_t stream) {
    (void)in_sizes; (void)n_in; (void)out_size; (void)d_ws; (void)ws_size;
    const int*   A1 = (const int*)  d_in[0];
    const int*   A2 = (const int*)  d_in[1];
    const int*   l1 = (const int*)  d_in[2];
    const int*   l2 = (const int*)  d_in[3];
    const float* nw = (const float*)d_in[4];
    const float* ew = (const float*)d_in[5];
    float* out = (float*)d_out;
    // Single persistent workgroup: whole problem (~95 KB state) lives in one
    // WGP's 320 KB LDS; Sinkhorn's sequential iterations sync with workgroup
    // barriers; 8 waves cooperate on the WMMA GEMM tiles.
    ged_kernel<<<1, BLOCK, 0, stream>>>(A1, A2, l1, l2, nw, ew, out);
}